// ECN4_37391985279550
// MI455X (gfx1250) — compile-verified
//
#include <hip/hip_runtime.h>
#include <math.h>

typedef __attribute__((ext_vector_type(2))) float v2f;
typedef __attribute__((ext_vector_type(4))) float v4f;
typedef __attribute__((ext_vector_type(8))) float v8f;

#define NNODES 8192
#define NGRAPH 64
#define KNN 3
#define NEDGE (NNODES*KNN)   // 24576
#define BNEPS 1e-5f
#define BIGF 1e30f

// ---------------- helpers ----------------

__device__ __forceinline__ void top3_ins(float d, int j,
                                         float& d0, float& d1, float& d2,
                                         int& j0, int& j1, int& j2) {
  if (d < d2) {
    if (d < d1) {
      if (d < d0) { d2=d1; j2=j1; d1=d0; j1=j0; d0=d; j0=j; }
      else        { d2=d1; j2=j1; d1=d;  j1=j; }
    } else        { d2=d;  j2=j; }
  }
}

// ---------------- graph ranges (batch is sorted) ----------------

__global__ void grange_init_k(int* gs, int* ge) {
  int i = threadIdx.x;
  if (i < NGRAPH) { gs[i] = 0; ge[i] = 0; }
}

__global__ void grange_k(const int* __restrict__ batch, int* gs, int* ge) {
  int i = blockIdx.x * blockDim.x + threadIdx.x;
  if (i >= NNODES) return;
  int b = batch[i];
  if (i == 0) gs[b] = 0;
  else if (batch[i-1] != b) { gs[b] = i; ge[batch[i-1]] = i; }
  if (i == NNODES-1) ge[b] = NNODES;
}

// ---------------- kNN on pos (3-dim, brute force) ----------------

__global__ void knn_pos_k(const float* __restrict__ pos,
                          const int* __restrict__ batch,
                          int* __restrict__ idx) {
  int i = blockIdx.x * blockDim.x + threadIdx.x;
  if (i >= NNODES) return;
  float px = pos[3*i], py = pos[3*i+1], pz = pos[3*i+2];
  int bi = batch[i];
  float d0=BIGF, d1=BIGF, d2=BIGF; int j0=i, j1=i, j2=i;
  for (int j = 0; j < NNODES; ++j) {
    if (j == i) continue;
    if (batch[j] != bi) continue;
    float dx = pos[3*j]   - px;
    float dy = pos[3*j+1] - py;
    float dz = pos[3*j+2] - pz;
    float d = dx*dx + dy*dy + dz*dz;
    top3_ins(d, j, d0, d1, d2, j0, j1, j2);
  }
  idx[3*i] = j0; idx[3*i+1] = j1; idx[3*i+2] = j2;
}

// ---------------- WMMA-based kNN on C-dim features ----------------
// 16 rows per block, A rows staged in padded LDS (conflict-free ds_load_b64
// fragment reads), B double-buffered so WMMA overlaps the next loads.

__global__ void knn_feat_k(const float* __restrict__ feat, int C,
                           const float* __restrict__ sq,
                           const int* __restrict__ batch,
                           int* __restrict__ idxout) {
  __shared__ float ldsA[16 * 260];        // 16 x (C+4), C <= 256
  __shared__ float dots[16 * 128];
  __shared__ float cd[256 * 3];
  __shared__ int   cj[256 * 3];
  int tid = threadIdx.x, wave = tid >> 5, lane = tid & 31;
  int rowBase = blockIdx.x * 16;
  int lstride = C + 4;

  // cooperative stage of the 16 A rows (each wave: 2 rows)
  for (int rr = 0; rr < 2; ++rr) {
    int r = wave * 2 + rr;
    const float* src = feat + (size_t)(rowBase + r) * C;
    float* dst = ldsA + r * lstride;
    for (int c = lane * 4; c < C; c += 128)
      *(v4f*)(dst + c) = *(const v4f*)(src + c);
  }
  __syncthreads();

  int myrow = tid & 15;       // row this thread scans
  int chunk = tid >> 4;       // 0..15, 8-column sub-chunk
  int inode = rowBase + myrow;
  int bi  = batch[inode];
  float sqi = sq[inode];
  float d0=BIGF, d1=BIGF, d2=BIGF; int j0=inode, j1=inode, j2=inode;

  int kb   = (lane >> 4) * 2;             // frag K pair per half-wave
  int colL = wave*16 + (lane & 15);       // this wave's tile column
  int rb   = (lane >> 4) * 8;             // D frag row base
  const float* ldsRow = ldsA + (lane & 15) * lstride;

  for (int cs = 0; cs < NNODES; cs += 128) {
    const float* Brow = feat + (size_t)(cs + colL) * C;  // B = feat^T, contiguous
    if (cs + 128 < NNODES)
      __builtin_prefetch(feat + (size_t)(cs + 128 + colL) * C, 0, 1);
    v8f acc = {};
    v2f bcur = *(const v2f*)(Brow + kb);
    for (int k = 0; k < C; k += 4) {
      int kn = (k + 4 < C) ? (k + 4) : k;
      v2f bnxt = *(const v2f*)(Brow + kn + kb);          // next-step B in flight
      v2f a    = *(const v2f*)(ldsRow + k + kb);         // ds_load_b64, no conflicts
      acc = __builtin_amdgcn_wmma_f32_16x16x4_f32(false, a, false, bcur,
                                                  (short)0, acc, false, false);
      bcur = bnxt;
    }
#pragma unroll
    for (int r = 0; r < 8; ++r) dots[(rb + r)*128 + colL] = acc[r];
    __syncthreads();
    int cbase = chunk * 8;
#pragma unroll
    for (int cc = 0; cc < 8; ++cc) {
      int j = cs + cbase + cc;
      if (j == inode) continue;
      if (batch[j] != bi) continue;
      float d = sqi + sq[j] - 2.0f * dots[myrow*128 + cbase + cc];
      top3_ins(d, j, d0, d1, d2, j0, j1, j2);
    }
    __syncthreads();
  }
  cd[tid*3+0]=d0; cd[tid*3+1]=d1; cd[tid*3+2]=d2;
  cj[tid*3+0]=j0; cj[tid*3+1]=j1; cj[tid*3+2]=j2;
  __syncthreads();
  if (tid < 16) {
    float e0=BIGF, e1=BIGF, e2=BIGF; int m0=inode, m1=inode, m2=inode;
    for (int t = tid; t < 256; t += 16)
      for (int s = 0; s < 3; ++s)
        top3_ins(cd[t*3+s], cj[t*3+s], e0, e1, e2, m0, m1, m2);
    idxout[(size_t)inode*3+0] = m0;
    idxout[(size_t)inode*3+1] = m1;
    idxout[(size_t)inode*3+2] = m2;
  }
}

// ---------------- squared norms ----------------

__global__ void sqnorm_k(const float* __restrict__ x, float* __restrict__ sq, int C) {
  int i = blockIdx.x * blockDim.x + threadIdx.x;
  if (i >= NNODES) return;
  float s = 0.f;
  for (int c = 0; c < C; ++c) { float v = x[(size_t)i*C + c]; s += v*v; }
  sq[i] = s;
}

// ---------------- edge feature gathers ----------------

__global__ void gather1_k(const float* __restrict__ x,
                          const int* __restrict__ idx,
                          float* __restrict__ e) {
  int t = blockIdx.x * blockDim.x + threadIdx.x;
  if (t >= NEDGE * 128) return;
  int c = t & 127; int eidx = t >> 7;
  int i = eidx / KNN; int j = idx[eidx];
  float v;
  if      (c <  59) v = x[i*59 + c];
  else if (c < 118) { int cc = c - 59; v = x[j*59 + cc] - x[i*59 + cc]; }
  else              v = 0.f;
  e[t] = v;
}

__global__ void gatherw_k(const float* __restrict__ xin,
                          const int* __restrict__ idx,
                          float* __restrict__ e, int C) {
  int t = blockIdx.x * blockDim.x + threadIdx.x;
  int C2 = 2 * C;
  if (t >= NEDGE * C2) return;
  int c = t % C2; int eidx = t / C2;
  int i = eidx / KNN; int j = idx[eidx];
  float v = (c < C) ? xin[(size_t)i*C + c]
                    : (xin[(size_t)j*C + (c - C)] - xin[(size_t)i*C + (c - C)]);
  e[t] = v;
}

__global__ void padw_k(const float* __restrict__ W, float* __restrict__ Wp) {
  int t = blockIdx.x * blockDim.x + threadIdx.x;
  if (t >= 128*128) return;
  int r = t >> 7;
  Wp[t] = (r < 118) ? W[t] : 0.f;
}

// ---------------- WMMA GEMM: Y = relu(A[M,Kd] @ W[Kd,N] + b) ----------------
// 256 threads = 8 waves; each wave owns NT=N/128 16x16 tiles so one block
// covers the full output width. A tile staged in padded LDS (shared by all
// 8 waves); per-k B fragments double-buffered; NT accumulators give NT
// back-to-back WMMAs per A fragment.

template<int NT>
__global__ void gemm_k(const float* __restrict__ A, const float* __restrict__ W,
                       const float* __restrict__ bias, float* __restrict__ Y,
                       int N, int Kd, int relu) {
  extern __shared__ float ldsA[];          // 16 x (Kd+4)
  int tid = threadIdx.x, wave = tid >> 5, lane = tid & 31;
  int rowBase = blockIdx.x * 16;
  int lstride = Kd + 4;

  for (int rr = 0; rr < 2; ++rr) {
    int r = wave * 2 + rr;
    const float* src = A + (size_t)(rowBase + r) * Kd;
    float* dst = ldsA + r * lstride;
    for (int c = lane * 4; c < Kd; c += 128)
      *(v4f*)(dst + c) = *(const v4f*)(src + c);
  }
  __syncthreads();

  int kb  = (lane >> 4) * 2;
  int col = wave * (NT*16) + (lane & 15);
  const float* Wc = W + col;
  const float* ldsRow = ldsA + (lane & 15) * lstride;

  v8f zero = {};
  v8f acc[NT];
#pragma unroll
  for (int t = 0; t < NT; ++t) acc[t] = zero;

  v2f bcur[NT], bnxt[NT];
#pragma unroll
  for (int t = 0; t < NT; ++t) {
    bcur[t].x = Wc[(size_t)(kb    ) * N + t*16];
    bcur[t].y = Wc[(size_t)(kb + 1) * N + t*16];
  }
  for (int k = 0; k < Kd; k += 4) {
    int kn = (k + 4 < Kd) ? (k + 4) : k;
#pragma unroll
    for (int t = 0; t < NT; ++t) {                       // next-step B in flight
      bnxt[t].x = Wc[(size_t)(kn + kb    ) * N + t*16];
      bnxt[t].y = Wc[(size_t)(kn + kb + 1) * N + t*16];
    }
    v2f a = *(const v2f*)(ldsRow + k + kb);              // ds_load_b64
#pragma unroll
    for (int t = 0; t < NT; ++t)
      acc[t] = __builtin_amdgcn_wmma_f32_16x16x4_f32(false, a, false, bcur[t],
                                                     (short)0, acc[t], false, false);
#pragma unroll
    for (int t = 0; t < NT; ++t) bcur[t] = bnxt[t];
  }

  int rbb = (lane >> 4) * 8;
#pragma unroll
  for (int t = 0; t < NT; ++t) {
    float bv = bias[col + t*16];
#pragma unroll
    for (int r = 0; r < 8; ++r) {
      float v = acc[t][r] + bv;
      if (relu) v = fmaxf(v, 0.f);
      Y[(size_t)(rowBase + rbb + r) * N + col + t*16] = v;
    }
  }
}

// ---------------- BatchNorm stats + apply ----------------

__global__ void colstats_k(const float* __restrict__ Y, int M, int N,
                           float* __restrict__ mean, float* __restrict__ rstd) {
  int c = blockIdx.x * 32 + threadIdx.x;   // blockDim (32,8)
  float s = 0.f, ss = 0.f;
  for (int r = threadIdx.y; r < M; r += 8) {
    float v = Y[(size_t)r * N + c];
    s += v; ss += v*v;
  }
  __shared__ float shs[8][32];
  __shared__ float shq[8][32];
  shs[threadIdx.y][threadIdx.x] = s;
  shq[threadIdx.y][threadIdx.x] = ss;
  __syncthreads();
  if (threadIdx.y == 0) {
    for (int y = 1; y < 8; ++y) { s += shs[y][threadIdx.x]; ss += shq[y][threadIdx.x]; }
    float inv = 1.f / (float)M;
    float m = s * inv;
    float var = fmaxf(ss * inv - m*m, 0.f);
    mean[c] = m;
    rstd[c] = rsqrtf(var + BNEPS);
  }
}

__global__ void bn_k(const float* __restrict__ Yin, float* __restrict__ Yout,
                     const float* __restrict__ mean, const float* __restrict__ rstd,
                     const float* __restrict__ g, const float* __restrict__ be,
                     const float* __restrict__ res, int total, int N) {
  int i = blockIdx.x * blockDim.x + threadIdx.x;
  if (i >= total) return;
  int c = i % N;
  float v = g[c] * (Yin[i] - mean[c]) * rstd[c] + be[c];
  if (res) v += res[i];
  Yout[i] = v;
}

// ---------------- mean over K=3 edges ----------------

__global__ void emean_k(const float* __restrict__ msg, float* __restrict__ out, int C) {
  int t = blockIdx.x * blockDim.x + threadIdx.x;
  if (t >= NNODES * C) return;
  int c = t % C; int i = t / C;
  float s = msg[(size_t)(i*KNN + 0)*C + c]
          + msg[(size_t)(i*KNN + 1)*C + c]
          + msg[(size_t)(i*KNN + 2)*C + c];
  out[t] = s * (1.f/3.f);
}

// ---------------- deterministic segment-mean pool (512-dim) ----------------

__global__ void pool_k(const float* __restrict__ x,
                       const int* __restrict__ gs, const int* __restrict__ ge,
                       float* __restrict__ pooled) {
  int g = blockIdx.x;                 // 64 blocks, 256 threads
  int s = gs[g], e = ge[g];
  float cnt = fmaxf((float)(e - s), 1.f);
  for (int c = threadIdx.x; c < 512; c += 256) {
    float acc = 0.f;
    for (int n = s; n < e; ++n) acc += x[(size_t)n * 512 + c];
    pooled[g*512 + c] = acc / cnt;
  }
}

// ---------------- cls2: 512->1, relu, BN over 64, sigmoid ----------------

__global__ void cls2_k(const float* __restrict__ h, const float* __restrict__ W,
                       const float* __restrict__ b, const float* __restrict__ g,
                       const float* __restrict__ be, float* __restrict__ out) {
  int r = threadIdx.x;  // 64 threads, one per graph
  float s = b[0];
  for (int k = 0; k < 512; ++k) s += h[(size_t)r*512 + k] * W[k];
  s = fmaxf(s, 0.f);
  __shared__ float sh[64];
  __shared__ float mv[2];
  sh[r] = s;
  __syncthreads();
  if (r == 0) {
    float m = 0.f;
    for (int i = 0; i < 64; ++i) m += sh[i];
    m *= (1.f/64.f);
    float v = 0.f;
    for (int i = 0; i < 64; ++i) { float d = sh[i] - m; v += d*d; }
    v *= (1.f/64.f);
    mv[0] = m; mv[1] = rsqrtf(v + BNEPS);
  }
  __syncthreads();
  float z = g[0] * (s - mv[0]) * mv[1] + be[0];
  out[r] = 1.f / (1.f + expf(-z));
}

// ---------------- host orchestration ----------------

extern "C" void kernel_launch(void* const* d_in, const int* in_sizes, int n_in,
                              void* d_out, int out_size, void* d_ws, size_t ws_size,
                              hipStream_t stream) {
  const float* x     = (const float*)d_in[0];
  const float* pos   = (const float*)d_in[1];
  const int*   batch = (const int*)d_in[2];
  const float* P[48];
  for (int i = 0; i < 48 && i < n_in; ++i) P[i] = (const float*)d_in[i];
  float* out = (float*)d_out;

  // workspace (requires ~181 MB): 3 edge buffers + node feats + scratch
  float* F = (float*)d_ws;
  size_t o = 0;
  auto alloc = [&](size_t n) { float* p = F + o; o += n; return p; };
  float* eA  = alloc((size_t)NEDGE * 512);
  float* eB  = alloc((size_t)NEDGE * 512);
  float* eC  = alloc((size_t)NEDGE * 512);
  float* x1a = alloc((size_t)NNODES * 128);
  float* x1b = alloc((size_t)NNODES * 256);
  float* x1c = alloc((size_t)NNODES * 512);
  float* Wp  = alloc(128 * 128);
  float* sq  = alloc(NNODES);
  float* cm  = alloc(512);
  float* cr  = alloc(512);
  float* pooled = alloc(64 * 512);
  float* h1  = alloc(64 * 512);
  int* idxb  = (int*)alloc(NNODES * KNN);
  int* gs    = (int*)alloc(64);
  int* ge    = (int*)alloc(64);

  auto nb = [](long n, int b) { return (int)((n + b - 1) / b); };
  size_t lds128 = (size_t)16 * (128 + 4) * sizeof(float);
  size_t lds256 = (size_t)16 * (256 + 4) * sizeof(float);
  size_t lds512 = (size_t)16 * (512 + 4) * sizeof(float);

  grange_init_k<<<1, 64, 0, stream>>>(gs, ge);
  grange_k<<<nb(NNODES,256), 256, 0, stream>>>(batch, gs, ge);

  // ===== conv1: knn(pos), edge MLP 118->128 + resmlp(2x128) =====
  knn_pos_k<<<nb(NNODES,256), 256, 0, stream>>>(pos, batch, idxb);
  padw_k<<<nb(128*128,256), 256, 0, stream>>>(P[4], Wp);
  gather1_k<<<nb((long)NEDGE*128,256), 256, 0, stream>>>(x, idxb, eA);
  gemm_k<1><<<NEDGE/16, 256, lds128, stream>>>(eA, Wp, P[5], eB, 128, 128, 1);
  colstats_k<<<4, dim3(32,8), 0, stream>>>(eB, NEDGE, 128, cm, cr);
  bn_k<<<nb((long)NEDGE*128,256), 256, 0, stream>>>(eB, eB, cm, cr, P[6], P[7], nullptr, NEDGE*128, 128);
  gemm_k<1><<<NEDGE/16, 256, lds128, stream>>>(eB, P[8], P[9], eC, 128, 128, 1);
  colstats_k<<<4, dim3(32,8), 0, stream>>>(eC, NEDGE, 128, cm, cr);
  bn_k<<<nb((long)NEDGE*128,256), 256, 0, stream>>>(eC, eC, cm, cr, P[10], P[11], nullptr, NEDGE*128, 128);
  gemm_k<1><<<NEDGE/16, 256, lds128, stream>>>(eC, P[12], P[13], eA, 128, 128, 1);
  colstats_k<<<4, dim3(32,8), 0, stream>>>(eA, NEDGE, 128, cm, cr);
  bn_k<<<nb((long)NEDGE*128,256), 256, 0, stream>>>(eA, eA, cm, cr, P[14], P[15], eB, NEDGE*128, 128);
  emean_k<<<nb((long)NNODES*128,256), 256, 0, stream>>>(eA, x1a, 128);

  // ===== conv2: knn(x1a 128-d), resmlp(3x256), residual = edge input =====
  sqnorm_k<<<nb(NNODES,256), 256, 0, stream>>>(x1a, sq, 128);
  knn_feat_k<<<NNODES/16, 256, 0, stream>>>(x1a, 128, sq, batch, idxb);
  gatherw_k<<<nb((long)NEDGE*256,256), 256, 0, stream>>>(x1a, idxb, eA, 128);
  gemm_k<2><<<NEDGE/16, 256, lds256, stream>>>(eA, P[16], P[17], eB, 256, 256, 1);
  colstats_k<<<8, dim3(32,8), 0, stream>>>(eB, NEDGE, 256, cm, cr);
  bn_k<<<nb((long)NEDGE*256,256), 256, 0, stream>>>(eB, eB, cm, cr, P[18], P[19], nullptr, NEDGE*256, 256);
  gemm_k<2><<<NEDGE/16, 256, lds256, stream>>>(eB, P[20], P[21], eC, 256, 256, 1);
  colstats_k<<<8, dim3(32,8), 0, stream>>>(eC, NEDGE, 256, cm, cr);
  bn_k<<<nb((long)NEDGE*256,256), 256, 0, stream>>>(eC, eC, cm, cr, P[22], P[23], nullptr, NEDGE*256, 256);
  gemm_k<2><<<NEDGE/16, 256, lds256, stream>>>(eC, P[24], P[25], eB, 256, 256, 1);
  colstats_k<<<8, dim3(32,8), 0, stream>>>(eB, NEDGE, 256, cm, cr);
  bn_k<<<nb((long)NEDGE*256,256), 256, 0, stream>>>(eB, eB, cm, cr, P[26], P[27], eA, NEDGE*256, 256);
  emean_k<<<nb((long)NNODES*256,256), 256, 0, stream>>>(eB, x1b, 256);

  // ===== conv3: knn(x1b 256-d), resmlp(3x512) =====
  sqnorm_k<<<nb(NNODES,256), 256, 0, stream>>>(x1b, sq, 256);
  knn_feat_k<<<NNODES/16, 256, 0, stream>>>(x1b, 256, sq, batch, idxb);
  gatherw_k<<<nb((long)NEDGE*512,256), 256, 0, stream>>>(x1b, idxb, eA, 256);
  gemm_k<4><<<NEDGE/16, 256, lds512, stream>>>(eA, P[28], P[29], eB, 512, 512, 1);
  colstats_k<<<16, dim3(32,8), 0, stream>>>(eB, NEDGE, 512, cm, cr);
  bn_k<<<nb((long)NEDGE*512,256), 256, 0, stream>>>(eB, eB, cm, cr, P[30], P[31], nullptr, NEDGE*512, 512);
  gemm_k<4><<<NEDGE/16, 256, lds512, stream>>>(eB, P[32], P[33], eC, 512, 512, 1);
  colstats_k<<<16, dim3(32,8), 0, stream>>>(eC, NEDGE, 512, cm, cr);
  bn_k<<<nb((long)NEDGE*512,256), 256, 0, stream>>>(eC, eC, cm, cr, P[34], P[35], nullptr, NEDGE*512, 512);
  gemm_k<4><<<NEDGE/16, 256, lds512, stream>>>(eC, P[36], P[37], eB, 512, 512, 1);
  colstats_k<<<16, dim3(32,8), 0, stream>>>(eB, NEDGE, 512, cm, cr);
  bn_k<<<nb((long)NEDGE*512,256), 256, 0, stream>>>(eB, eB, cm, cr, P[38], P[39], eA, NEDGE*512, 512);
  emean_k<<<nb((long)NNODES*512,256), 256, 0, stream>>>(eB, x1c, 512);

  // ===== pool + classifier =====
  pool_k<<<NGRAPH, 256, 0, stream>>>(x1c, gs, ge, pooled);

  gemm_k<4><<<4, 256, lds512, stream>>>(pooled, P[40], P[41], h1, 512, 512, 1);
  colstats_k<<<16, dim3(32,8), 0, stream>>>(h1, 64, 512, cm, cr);
  bn_k<<<nb(64*512,256), 256, 0, stream>>>(h1, h1, cm, cr, P[42], P[43], nullptr, 64*512, 512);

  cls2_k<<<1, 64, 0, stream>>>(h1, P[44], P[45], P[46], P[47], out);
}